// CTC_64570538328649
// MI455X (gfx1250) — compile-verified
//
#include <hip/hip_runtime.h>
#include <stdint.h>

#define CCLS 256
#define TPB  544          // 17 wave32s: one DP state per thread (L = 2S+1 <= 544)
#define GRP  8            // timestep rows per async group (8 * 1KB = 8KB per buffer)
#define NEGV (-1e30f)

__device__ __forceinline__ uint32_t lds_addr_u32(const void* p) {
  // generic pointer to LDS: low 32 bits are the LDS byte offset
  return (uint32_t)(uintptr_t)p;
}

__device__ __forceinline__ float logadd3(float a, float b, float c) {
  float m = fmaxf(fmaxf(a, b), c);
  float s = __expf(a - m) + __expf(b - m) + __expf(c - m);
  return m + __logf(s);
}

// split workgroup barrier pieces
__device__ __forceinline__ void bar_signal_after_store() {
  asm volatile("s_wait_dscnt 0x0\n\ts_barrier_signal -1" ::: "memory");
}
__device__ __forceinline__ void bar_signal_after_store_and_async() {
  // group-final step: also guarantees the in-flight async group has landed,
  // so this step's barrier doubles as the visibility barrier for the next group
  asm volatile("s_wait_dscnt 0x0\n\ts_wait_asynccnt 0x0\n\ts_barrier_signal -1" ::: "memory");
}
__device__ __forceinline__ void bar_wait() {
  asm volatile("s_barrier_wait -1" ::: "memory");
}

__global__ __launch_bounds__(TPB)
void ctc_alpha_kernel(const float* __restrict__ log_probs,   // [T,B,C]
                      const int*   __restrict__ y,           // [B,S]
                      const int*   __restrict__ input_lengths,   // [B]
                      const int*   __restrict__ target_lengths,  // [B]
                      float* __restrict__ loss_ws,           // [B]
                      int T, int B, int S) {
  __shared__ float lp_sh[2][GRP][CCLS];   // 16 KB double-buffered row groups
  __shared__ float A[2][TPB + 4];         // alpha double buffer, 2 lead pads

  const int b   = blockIdx.x;
  const int tid = threadIdx.x;
  if (b >= B) return;

  const int L    = 2 * S + 1;
  const int ilen = input_lengths[b];      // block-uniform
  const int tlen = target_lengths[b];     // block-uniform

  // effective number of timesteps actually advancing alpha (uniform control flow)
  const int Teff = min(T, (ilen < 1) ? 1 : ilen);

  // ---- per-thread state metadata (exactly one state per thread) ----
  const int  s0   = tid;
  const bool act  = (s0 < L);
  int  e0 = 0;
  bool k0 = false;
  if (act && (s0 & 1)) {
    int j = (s0 - 1) >> 1;
    e0 = y[b * S + j];
    k0 = (s0 >= 3) && (e0 != y[b * S + j - 1]);
  }

  // ---- async copy setup: lanes 0..511 each DMA 16B of the 8KB group to LDS ----
  const int      r        = (tid < 512) ? (tid >> 6) : 0;   // row within group
  const int      c0       = (tid & 63) << 2;                // starting class
  const uint32_t voff0    = (uint32_t)((((uint64_t)r * B + b) * CCLS + c0) * 4u);
  const uint32_t grpStep  = (uint32_t)((uint64_t)GRP * B * CCLS * 4u);
  const uint32_t ldsDstA  = lds_addr_u32(&lp_sh[0][0][0]) + (uint32_t)tid * 16u;
  const uint32_t ldsDstB  = lds_addr_u32(&lp_sh[1][0][0]) + (uint32_t)tid * 16u;
  const int      ngroups  = (Teff + GRP - 1) / GRP;

  auto issue_group = [&](int k) {
    if (tid < 512 && (k * GRP + r) < T) {
      uint32_t dst = (k & 1) ? ldsDstB : ldsDstA;
      uint32_t off = voff0 + (uint32_t)k * grpStep;
      asm volatile("global_load_async_to_lds_b128 %0, %1, %2 offset:0"
                   :: "v"(dst), "v"(off), "s"(log_probs) : "memory");
    }
  };

  // Prime the pipeline: group 0 and (if present) group 1 in flight.
  issue_group(0);
  if (ngroups > 1) {
    issue_group(1);
    asm volatile("s_wait_asynccnt 1" ::: "memory");   // group 0 landed
  } else {
    asm volatile("s_wait_asynccnt 0" ::: "memory");
  }
  __syncthreads();                                    // group-0 data visible everywhere

  // ---- alpha init (t = 0) into A[1] ----
  if (tid == 0) { A[0][0] = NEGV; A[0][1] = NEGV; A[1][0] = NEGV; A[1][1] = NEGV; }
  if (act) {
    float v = NEGV;
    if (s0 == 0)      v = lp_sh[0][0][0];
    else if (s0 == 1) v = (tlen > 0) ? lp_sh[0][0][e0] : NEGV;
    A[1][s0 + 2] = v;
  }
  __syncthreads();

  // lp value for the *current* timestep lives in a register (prefetched under the
  // split barrier of the previous step, or at a group header).
  float lpv0 = 0.0f;

  // one DP step; lastInGroup folds the async-group wait into this step's barrier
  auto step = [&](int t, bool lastInGroup) {
    const float* __restrict__ AO = A[t & 1];
    float*       __restrict__ AN = A[(t + 1) & 1];

    float a  = AO[s0 + 2];
    float b1 = AO[s0 + 1];
    float c1 = k0 ? AO[s0] : NEGV;
    float nv = fmaxf(logadd3(a, b1, c1) + lpv0, NEGV);
    if (act) AN[s0 + 2] = nv;

    if (lastInGroup) bar_signal_after_store_and_async();
    else             bar_signal_after_store();

    const int tn = t + 1;
    if ((tn & (GRP - 1)) != 0 && tn < Teff) {   // next row in already-landed group
      lpv0 = lp_sh[(tn >> 3) & 1][tn & (GRP - 1)][e0];
    }
    bar_wait();
  };

  // ---- prologue: remaining rows of group 0 (t = 1..7) ----
  if (Teff > 1) {
    lpv0 = lp_sh[0][1][e0];
    for (int t = 1; t < GRP && t < Teff; ++t) step(t, t == GRP - 1);
  }

  // ---- main: one 8KB group per outer iteration, inner 8 steps unrolled ----
  for (int k = 1; k < ngroups; ++k) {
    if (k + 1 < ngroups) issue_group(k + 1);     // into buffer freed last barrier
    lpv0 = lp_sh[k & 1][0][e0];                  // group already landed (prev step)

    if (k * GRP + GRP <= Teff) {                 // full group: unguarded unroll
#pragma unroll
      for (int rr = 0; rr < GRP; ++rr) step(k * GRP + rr, rr == GRP - 1);
    } else {                                     // partial tail group
      for (int rr = 0; rr < GRP; ++rr) {
        const int t = k * GRP + rr;
        if (t < Teff) step(t, false);
      }
    }
  }

  // ---- finalize per batch element (last barrier synced everyone) ----
  if (tid == 0) {
    const float* AF = A[Teff & 1];
    const int last  = 2 * tlen;
    float ae = AF[last + 2];
    float ap = (tlen > 0) ? AF[last + 1] : NEGV;
    float m  = fmaxf(ae, ap);
    float nll = -(m + __logf(__expf(ae - m) + __expf(ap - m)));
    if (nll > 1e29f) nll = 0.0f;
    int denom = (tlen > 0) ? tlen : 1;
    loss_ws[b] = nll / (float)denom;
  }
}

__global__ void ctc_reduce_kernel(const float* __restrict__ ws,
                                  float* __restrict__ out, int B) {
  __shared__ float sm[256];
  const int tid = threadIdx.x;
  float v = 0.0f;
  for (int i = tid; i < B; i += blockDim.x) v += ws[i];
  sm[tid] = v;
  __syncthreads();
  for (int off = 128; off > 0; off >>= 1) {
    if (tid < off) sm[tid] += sm[tid + off];
    __syncthreads();
  }
  if (tid == 0) out[0] = sm[0] / (float)B;
}

extern "C" void kernel_launch(void* const* d_in, const int* in_sizes, int n_in,
                              void* d_out, int out_size, void* d_ws, size_t ws_size,
                              hipStream_t stream) {
  const float* log_probs      = (const float*)d_in[0];
  const int*   y              = (const int*)d_in[1];
  const int*   input_lengths  = (const int*)d_in[2];
  const int*   target_lengths = (const int*)d_in[3];

  const int B = in_sizes[2];                 // input_lengths has B elements
  const int S = in_sizes[1] / B;             // y is [B, S]
  const int T = in_sizes[0] / (B * CCLS);    // log_probs is [T, B, C]

  float* ws  = (float*)d_ws;                 // B partial losses
  float* out = (float*)d_out;

  ctc_alpha_kernel<<<B, TPB, 0, stream>>>(log_probs, y, input_lengths,
                                          target_lengths, ws, T, B, S);
  ctc_reduce_kernel<<<1, 256, 0, stream>>>(ws, out, B);
}